// TimeSwiGLU_67954972557422
// MI455X (gfx1250) — compile-verified
//
#include <hip/hip_runtime.h>
#include <math.h>

typedef __attribute__((ext_vector_type(2))) float v2f;
typedef __attribute__((ext_vector_type(8))) float v8f;

#define WMMA_F32(a, b, c) \
  __builtin_amdgcn_wmma_f32_16x16x4_f32(false, (a), false, (b), (short)0, (c), false, false)

constexpr int Bb = 2;
constexpr int Ss = 2048;
constexpr int Hh = 12;
constexpr int Dd = 64;
constexpr int BH = Bb * Hh;        // 24 (b,h) pairs
constexpr int NCH = 16;            // S chunks
constexpr int CH = Ss / NCH;       // 128 rows per chunk
constexpr int SROW = 7 * Hh * Dd;  // 5376 floats between consecutive s rows
constexpr int MSZ = Dd * Dd;       // 4096 floats per 64x64 matrix

// qkv[b, s, slot, h, d]; slots: q1=0,k1=1,q2=2,k2=3,q3=4,k3=5,v=6
__device__ __forceinline__ size_t qoff(int b, int s, int slot, int h, int d) {
  return ((((size_t)b * Ss + s) * 7 + slot) * Hh + h) * (size_t)Dd + d;
}

// ---------------- Kernel 1: partial M1 = K1^T V, M2 = K2^T V ----------------
// 8 waves: waves 0-3 -> M1 strips, waves 4-7 -> M2 strips (disjoint outputs,
// no cross-wave reduction needed). Each wave: 4 x v8f accumulators = 16x64.
__global__ void __launch_bounds__(256) ktv_partials(const float* __restrict__ qkv,
                                                    float* __restrict__ P1,
                                                    float* __restrict__ P2) {
  const int bh = blockIdx.x / NCH, ch = blockIdx.x % NCH;
  const int b = bh / Hh, h = bh % Hh;
  const int wave = threadIdx.x >> 5, lane = threadIdx.x & 31;
  const int mat = wave >> 2;          // 0: M1 (k1), 1: M2 (k2)
  const int irow = wave & 3;          // i-tile (rows of K^T)
  const int kslot = mat ? 3 : 1;
  const int m = lane & 15;            // A: M index / B: N index
  const int kk = (lane >> 4) << 1;    // K sub-offset per half-wave
  v8f acc[4] = {};
  const int s0b = ch * CH;
  for (int s0 = s0b; s0 < s0b + CH; s0 += 4) {
    const float* K0 = qkv + qoff(b, s0 + kk, kslot, h, 0);
    const float* V0 = qkv + qoff(b, s0 + kk, 6, h, 0);
    v2f a;
    a.x = K0[irow * 16 + m];          // A[m][kk]   = K[s0+kk][i]
    a.y = K0[SROW + irow * 16 + m];   // A[m][kk+1] = K[s0+kk+1][i]
#pragma unroll
    for (int j = 0; j < 4; ++j) {
      v2f bt;
      bt.x = V0[j * 16 + m];          // B[kk][n]   = V[s0+kk][j]
      bt.y = V0[SROW + j * 16 + m];   // B[kk+1][n]
      acc[j] = WMMA_F32(a, bt, acc[j]);
    }
  }
  float* P = (mat ? P2 : P1) + ((size_t)bh * NCH + ch) * MSZ;
  const int rh = (lane >> 4) * 8;     // C/D: lanes 16-31 hold rows M+8
#pragma unroll
  for (int j = 0; j < 4; ++j)
#pragma unroll
    for (int r = 0; r < 8; ++r)
      P[(irow * 16 + rh + r) * Dd + j * 16 + m] = acc[j][r];
}

// ---------------- deterministic chunk-partial reduction ----------------
__global__ void __launch_bounds__(256) reduce_partials(const float* __restrict__ P,
                                                       float* __restrict__ M) {
  const int bh = blockIdx.x;
  for (int e = threadIdx.x; e < MSZ; e += 256) {
    float s = 0.f;
#pragma unroll
    for (int c = 0; c < NCH; ++c) s += P[((size_t)bh * NCH + c) * MSZ + e];
    M[(size_t)bh * MSZ + e] = s;
  }
}

// ---------------- Kernel 2: u = silu(Q1 M1) * (Q2 M2), u[bh][s][64] ----------------
__global__ void __launch_bounds__(256) u_kernel(const float* __restrict__ qkv,
                                                const float* __restrict__ M1,
                                                const float* __restrict__ M2,
                                                float* __restrict__ U) {
  __shared__ float sM1[MSZ];
  __shared__ float sM2[MSZ];
  const int bh = blockIdx.x / NCH, ch = blockIdx.x % NCH;
  const int b = bh / Hh, h = bh % Hh;
  for (int e = threadIdx.x; e < MSZ; e += 256) {
    sM1[e] = M1[(size_t)bh * MSZ + e];
    sM2[e] = M2[(size_t)bh * MSZ + e];
  }
  __syncthreads();
  const int wave = threadIdx.x >> 5, lane = threadIdx.x & 31;
  const int tbase = ch * CH + wave * 16;   // 16 t-rows per wave
  const int m = lane & 15, kk = (lane >> 4) << 1;
  v8f acc1[4] = {}, acc2[4] = {};
  for (int step = 0; step < 16; ++step) {  // K = 64 inner dim, 4 per WMMA
    const int k0 = step * 4 + kk;
    const float* q1p = qkv + qoff(b, tbase + m, 0, h, k0);
    const float* q2p = qkv + qoff(b, tbase + m, 2, h, k0);
    v2f a1, a2;
    a1.x = q1p[0]; a1.y = q1p[1];
    a2.x = q2p[0]; a2.y = q2p[1];
#pragma unroll
    for (int j = 0; j < 4; ++j) {
      v2f b1, b2;
      b1.x = sM1[k0 * Dd + j * 16 + m];
      b1.y = sM1[(k0 + 1) * Dd + j * 16 + m];
      b2.x = sM2[k0 * Dd + j * 16 + m];
      b2.y = sM2[(k0 + 1) * Dd + j * 16 + m];
      acc1[j] = WMMA_F32(a1, b1, acc1[j]);
      acc2[j] = WMMA_F32(a2, b2, acc2[j]);
    }
  }
  const int rh = (lane >> 4) * 8;
#pragma unroll
  for (int j = 0; j < 4; ++j)
#pragma unroll
    for (int r = 0; r < 8; ++r) {
      const float x = acc1[j][r];
      const float t1 = x / (1.f + __expf(-x));   // silu
      U[((size_t)bh * Ss + tbase + rh + r) * Dd + j * 16 + m] = t1 * acc2[j][r];
    }
}

// ---------------- Kernel 3: partial M3 = K3^T u ----------------
__global__ void __launch_bounds__(128) ktu_partials(const float* __restrict__ qkv,
                                                    const float* __restrict__ U,
                                                    float* __restrict__ P3) {
  const int bh = blockIdx.x / NCH, ch = blockIdx.x % NCH;
  const int b = bh / Hh, h = bh % Hh;
  const int wave = threadIdx.x >> 5, lane = threadIdx.x & 31;
  const int irow = wave;
  const int m = lane & 15, kk = (lane >> 4) << 1;
  v8f acc[4] = {};
  const int s0b = ch * CH;
  for (int s0 = s0b; s0 < s0b + CH; s0 += 4) {
    const float* K0 = qkv + qoff(b, s0 + kk, 5, h, 0);
    const float* U0 = U + ((size_t)bh * Ss + s0 + kk) * Dd;
    v2f a;
    a.x = K0[irow * 16 + m];
    a.y = K0[SROW + irow * 16 + m];
#pragma unroll
    for (int j = 0; j < 4; ++j) {
      v2f bt;
      bt.x = U0[j * 16 + m];
      bt.y = U0[Dd + j * 16 + m];
      acc[j] = WMMA_F32(a, bt, acc[j]);
    }
  }
  float* P = P3 + ((size_t)bh * NCH + ch) * MSZ;
  const int rh = (lane >> 4) * 8;
#pragma unroll
  for (int j = 0; j < 4; ++j)
#pragma unroll
    for (int r = 0; r < 8; ++r)
      P[(irow * 16 + rh + r) * Dd + j * 16 + m] = acc[j][r];
}

// ---------------- Kernel 4: out = Q3 M3, out[b][t][h][d] ----------------
__global__ void __launch_bounds__(256) out_kernel(const float* __restrict__ qkv,
                                                  const float* __restrict__ M3,
                                                  float* __restrict__ out) {
  __shared__ float sM[MSZ];
  const int bh = blockIdx.x / NCH, ch = blockIdx.x % NCH;
  const int b = bh / Hh, h = bh % Hh;
  for (int e = threadIdx.x; e < MSZ; e += 256) sM[e] = M3[(size_t)bh * MSZ + e];
  __syncthreads();
  const int wave = threadIdx.x >> 5, lane = threadIdx.x & 31;
  const int tbase = ch * CH + wave * 16;
  const int m = lane & 15, kk = (lane >> 4) << 1;
  v8f acc[4] = {};
  for (int step = 0; step < 16; ++step) {
    const int k0 = step * 4 + kk;
    const float* qp = qkv + qoff(b, tbase + m, 4, h, k0);
    v2f a; a.x = qp[0]; a.y = qp[1];
#pragma unroll
    for (int j = 0; j < 4; ++j) {
      v2f bt;
      bt.x = sM[k0 * Dd + j * 16 + m];
      bt.y = sM[(k0 + 1) * Dd + j * 16 + m];
      acc[j] = WMMA_F32(a, bt, acc[j]);
    }
  }
  const int rh = (lane >> 4) * 8;
#pragma unroll
  for (int j = 0; j < 4; ++j)
#pragma unroll
    for (int r = 0; r < 8; ++r)
      out[(((size_t)b * Ss + tbase + rh + r) * Hh + h) * Dd + j * 16 + m] = acc[j][r];
}

extern "C" void kernel_launch(void* const* d_in, const int* in_sizes, int n_in,
                              void* d_out, int out_size, void* d_ws, size_t ws_size,
                              hipStream_t stream) {
  (void)in_sizes; (void)n_in; (void)out_size; (void)ws_size;
  const float* qkv = (const float*)d_in[0];
  float* out = (float*)d_out;
  float* w = (float*)d_ws;

  const size_t PSZ = (size_t)BH * NCH * MSZ;  // 1,572,864 floats per partial set
  const size_t MTOT = (size_t)BH * MSZ;       // 98,304 floats per 64x64 set
  float* P1 = w;
  float* P2 = P1 + PSZ;
  float* P3 = P2 + PSZ;
  float* M1 = P3 + PSZ;
  float* M2 = M1 + MTOT;
  float* M3 = M2 + MTOT;
  float* U  = M3 + MTOT;                      // BH*S*D = 3,145,728 floats

  ktv_partials<<<BH * NCH, 256, 0, stream>>>(qkv, P1, P2);
  reduce_partials<<<BH, 256, 0, stream>>>(P1, M1);
  reduce_partials<<<BH, 256, 0, stream>>>(P2, M2);
  u_kernel<<<BH * NCH, 256, 0, stream>>>(qkv, M1, M2, U);
  ktu_partials<<<BH * NCH, 128, 0, stream>>>(qkv, U, P3);
  reduce_partials<<<BH, 256, 0, stream>>>(P3, M3);
  out_kernel<<<BH * NCH, 256, 0, stream>>>(qkv, M3, out);
}